// NormalMixtureEM_48378511622819
// MI455X (gfx1250) — compile-verified
//
#include <hip/hip_runtime.h>
#include <hip/hip_bf16.h>
#include <math.h>

// NormalMixtureEM for MI455X (gfx1250): one block per batch row.
// 512 threads = 16 wave32s; one window sample per thread held in a register.
// Async global->LDS (ASYNCcnt) staging + v_wmma_f32_16x16x4_f32 cross-wave
// reduction are the CDNA5-specific paths.

#define EM_L 512
#define EM_K 4
#define EM_ITERS 5
#define EM_EPS 1e-8f
#define EM_NOISE 0.01f

typedef __attribute__((ext_vector_type(2))) float v2f;
typedef __attribute__((ext_vector_type(8))) float v8f;
typedef int em_v4i __attribute__((vector_size(4 * sizeof(int))));

#if defined(__gfx1250__)
#define EM_GFX1250 1
#endif

__device__ __forceinline__ float em_sigmoid(float x) {
  return 1.0f / (1.0f + __expf(-x));
}

// Reduce 12 per-thread values across a 512-thread block into totals[0..11].
// Intra-wave: xor-shuffle tree. Cross-wave: wave 0 sums the 16 per-wave
// partial rows with 4 chained v_wmma_f32_16x16x4_f32 (A = all ones, so
// D[m,n] = sum_j B[j,n] + C[m,n]; column sums are invariant to the exact
// B row->VGPR mapping, only the lane<->column mapping matters).
__device__ __forceinline__ void block_reduce12(float v[12], float (*part)[16],
                                               float* totals, int lane, int wave) {
#pragma unroll
  for (int m = 16; m >= 1; m >>= 1) {
#pragma unroll
    for (int q = 0; q < 12; ++q) v[q] += __shfl_xor(v[q], m, 32);
  }
  if (lane == 0) {
#pragma unroll
    for (int q = 0; q < 12; ++q) part[wave][q] = v[q];
#pragma unroll
    for (int q = 12; q < 16; ++q) part[wave][q] = 0.0f;
  }
  __syncthreads();
  if (wave == 0) {  // wave-uniform branch: EXEC stays all-1s inside (WMMA req.)
#if defined(EM_GFX1250) && __has_builtin(__builtin_amdgcn_wmma_f32_16x16x4_f32)
    v2f a;
    a.x = 1.0f;
    a.y = 1.0f;
    v8f c = {};
    const int col = lane & 15;
    const int half = (lane < 16) ? 0 : 2;
#pragma unroll
    for (int i = 0; i < 4; ++i) {
      v2f bv;
      bv.x = part[4 * i + half + 0][col];
      bv.y = part[4 * i + half + 1][col];
      // (neg_a, A, neg_b, B, c_mod, C, reuse_a, reuse_b)
      c = __builtin_amdgcn_wmma_f32_16x16x4_f32(false, a, false, bv, (short)0, c,
                                                false, false);
    }
    // C/D layout: VGPR0, lanes 0-15 hold row M=0, cols = lane.
    if (lane < 12) totals[lane] = c[0];
#else
    if (lane < 12) {
      float s = 0.0f;
#pragma unroll
      for (int wv = 0; wv < 16; ++wv) s += part[wv][lane];
      totals[lane] = s;
    }
#endif
  }
  __syncthreads();
}

__global__ __launch_bounds__(EM_L) void NormalMixtureEM_48378511622819_kernel(
    const float* __restrict__ windows,        // [B, L]
    const float* __restrict__ init_centers,   // [K]
    const float* __restrict__ init_scales,    // [K]
    const float* __restrict__ init_weights,   // [K]
    const float* __restrict__ prior_p_param,  // [1]
    const float* __restrict__ weights_param,  // [1, L]
    const float* __restrict__ blend,          // [1]
    const float* __restrict__ noise,          // [B, K]
    float* __restrict__ out,                  // g | w | mu | sigma (flat)
    int nB) {
  __shared__ float xrow[EM_L];
  __shared__ float part[16][16];
  __shared__ float totals[16];

  const int b = blockIdx.x;
  const int t = threadIdx.x;
  const int lane = t & 31;
  const int wave = t >> 5;

  // ---- Stage the window row into LDS with async global->LDS B128 loads ----
  const float* grow = windows + (size_t)b * EM_L;
#if defined(EM_GFX1250) && __has_builtin(__builtin_amdgcn_global_load_async_to_lds_b128)
  if (t < EM_L / 4) {
    float* gp = const_cast<float*>(grow) + t * 4;
    __builtin_amdgcn_global_load_async_to_lds_b128(
        (__attribute__((address_space(1))) em_v4i*)gp,
        (__attribute__((address_space(3))) em_v4i*)(&xrow[t * 4]),
        /*offset=*/0, /*cpol=*/0);
  }
#if __has_builtin(__builtin_amdgcn_s_wait_asynccnt)
  __builtin_amdgcn_s_wait_asynccnt(0);
#else
  asm volatile("s_wait_asynccnt 0" ::: "memory");
#endif
#else
  xrow[t] = grow[t];
#endif
  __syncthreads();

  const float x = xrow[t];
  const float wt = __expf(weights_param[t]);  // column weight (uniform per l)
  const float prior_p = em_sigmoid(prior_p_param[0]);
  const float blender = em_sigmoid(blend[0]);

  // ---- Row statistics: mean, unbiased var/std ----
  float v[12];
  v[0] = x;
  v[1] = x * x;
#pragma unroll
  for (int q = 2; q < 12; ++q) v[q] = 0.0f;
  block_reduce12(v, part, totals, lane, wave);
  const float Lf = (float)EM_L;
  const float Sx = totals[0];
  const float Sxx = totals[1];
  const float mean = Sx / Lf;
  float prior_var = (Sxx - Sx * Sx / Lf) * (1.0f / (Lf - 1.0f));
  prior_var = fmaxf(prior_var, 0.0f);
  const float dstd = sqrtf(prior_var);

  // ---- Initialize per-row mixture parameters (K=4, redundant per thread) ----
  float mu[EM_K], sig[EM_K], w[EM_K];
#pragma unroll
  for (int k = 0; k < EM_K; ++k) {
    mu[k] = init_centers[k] * dstd + mean +
            noise[(size_t)b * EM_K + k] * dstd * EM_NOISE;
    sig[k] = fabsf(init_scales[k]) * dstd;
    w[k] = init_weights[k];
  }

  float gk[EM_K] = {0.0f, 0.0f, 0.0f, 0.0f};

  for (int it = 0; it < EM_ITERS; ++it) {
    // ---- E-step: stable softmax over K per sample ----
    float inv_s[EM_K], lm[EM_K];
    float mmax = -3.0e38f;
#pragma unroll
    for (int k = 0; k < EM_K; ++k) {
      inv_s[k] = 1.0f / (sig[k] + EM_EPS);
      const float z = (x - mu[k]) * inv_s[k];
      lm[k] = __logf((w[k] + EM_EPS) * inv_s[k]) - 0.5f * z * z;
      mmax = fmaxf(mmax, lm[k]);
    }
    float es = 0.0f;
#pragma unroll
    for (int k = 0; k < EM_K; ++k) {
      gk[k] = __expf(lm[k] - mmax);
      es += gk[k];
    }
    const float inv_es = 1.0f / es;
#pragma unroll
    for (int k = 0; k < EM_K; ++k) {
      gk[k] *= inv_es;
      const float wg = gk[k] * wt;
      v[k] = wg;              // S0
      v[4 + k] = wg * x;      // S1
      v[8 + k] = wg * x * x;  // S2
    }
    block_reduce12(v, part, totals, lane, wave);

    // ---- M-step (redundant per thread; uses S2-2*mu*S1+mu^2*S0 identity) ----
    float wsum = 0.0f;
    float wnew[EM_K];
#pragma unroll
    for (int k = 0; k < EM_K; ++k) {
      const float S0 = totals[k];
      const float S1 = totals[4 + k];
      const float S2 = totals[8 + k];
      const float sg = fmaxf(S0, EM_EPS);
      wnew[k] = sg + prior_p;
      wsum += wnew[k];
      const float data_mu = S1 / sg;
      const float muk = data_mu * blender + mean * (1.0f - blender);
      const float data_var = (S2 - 2.0f * muk * S1 + muk * muk * S0) / sg;
      const float var = data_var * blender + prior_var * (1.0f - blender);
      mu[k] = muk;
      sig[k] = sqrtf(var + EM_EPS);
    }
    const float inv_wsum = 1.0f / wsum;
#pragma unroll
    for (int k = 0; k < EM_K; ++k) w[k] = wnew[k] * inv_wsum;
  }

  // ---- Outputs: g from the LAST E-step; w/mu/sigma after the last M-step ----
  float4 gout;
  gout.x = gk[0];
  gout.y = gk[1];
  gout.z = gk[2];
  gout.w = gk[3];
  ((float4*)out)[(size_t)b * EM_L + t] = gout;  // g[b, l, :] coalesced 16B

  if (t == 0) {
    const size_t gsz = (size_t)nB * EM_L * EM_K;
    const size_t bk = (size_t)nB * EM_K;
#pragma unroll
    for (int k = 0; k < EM_K; ++k) {
      out[gsz + (size_t)b * EM_K + k] = w[k];
      out[gsz + bk + (size_t)b * EM_K + k] = mu[k];
      out[gsz + 2 * bk + (size_t)b * EM_K + k] = sig[k];
    }
  }
}

extern "C" void kernel_launch(void* const* d_in, const int* in_sizes, int n_in,
                              void* d_out, int out_size, void* d_ws, size_t ws_size,
                              hipStream_t stream) {
  const float* windows = (const float*)d_in[0];
  const float* init_centers = (const float*)d_in[1];
  const float* init_scales = (const float*)d_in[2];
  const float* init_weights = (const float*)d_in[3];
  const float* prior_p_param = (const float*)d_in[4];
  const float* weights_param = (const float*)d_in[5];
  const float* blend = (const float*)d_in[6];
  const float* noise = (const float*)d_in[7];

  const int nB = in_sizes[0] / EM_L;

  hipLaunchKernelGGL(NormalMixtureEM_48378511622819_kernel, dim3(nB),
                     dim3(EM_L), 0, stream, windows, init_centers, init_scales,
                     init_weights, prior_p_param, weights_param, blend, noise,
                     (float*)d_out, nB);
}